// Generator_21629455302970
// MI455X (gfx1250) — compile-verified
//
#include <hip/hip_runtime.h>
#include <hip/hip_bf16.h>

// ===========================================================================
// 6-qubit VQC batch simulator for MI455X (gfx1250).
//
//  * Fold all p/weight-dependent gates per layer into dense 64x64 complex
//    operators F_l (23 layers) -> real 128x128 GEMM operators, built by
//    prep_kernel directly in WMMA B-operand-swizzled layout.
//  * Main kernel: state (16 elements x 128 f32 per wave) in LDS; per layer a
//    (16x128)x(128x128) f32 GEMM on V_WMMA_F32_16X16X4_F32 (fp32 WMMA keeps
//    precision over ~800 chained gates), then per-element data gates
//    (RY(x1)*RX(x0) on all 6 qubits) as VALU on the LDS state.
//  * G_l staging is software-pipelined through registers: global loads for
//    layer l+1 are issued before the layer-l GEMM and land during it.
//
// WMMA operand layouts (CDNA5 ISA 7.12.2, wave32):
//   A 16x4 f32 : VGPR0 = K=0 (lanes 0-15) / K=2 (lanes 16-31), VGPR1 = K=1/K=3
//   C/D 16x16  : VGPR r, lanes 0-15 -> (M=r, N=lane); lanes 16-31 -> (M=r+8)
//   B 4x16     : assumed mirror of A (lane half = K pair, VGPR = K within pair,
//                lane%16 = N).  G is pre-swizzled so one ds_load_b128 yields
//                the B operands of two N-tiles in even-aligned VGPR pairs.
// ===========================================================================

#define NQ   6
#define QDIM 64
#define NLAY 23

typedef __attribute__((ext_vector_type(2))) float v2f;
typedef __attribute__((ext_vector_type(8))) float v8f;

// B-swizzled index for G element (k=0..127, n=0..127):
//   kt=k>>2, hk=(k>>1)&1, kk=k&1, nt=n>>4, m=n&15, nt2=nt>>1, no=nt&1
//   idx = ((((kt*2+hk)*4 + nt2)*16 + m)*4) + no*2 + kk
__host__ __device__ inline int gbidx(int k, int n) {
    int kt = k >> 2, hk = (k >> 1) & 1, kk = k & 1;
    int nt = n >> 4, m = n & 15, nt2 = nt >> 1, no = nt & 1;
    return ((((kt * 2 + hk) * 4 + nt2) * 16 + m) << 2) + no * 2 + kk;
}

// ---------------------------------------------------------------------------
// Prep: build swizzled G_l (128x128 f32) for each layer into d_ws.
// One block, 64 threads; thread 'col' evolves identity column 'col' in LDS.
// ---------------------------------------------------------------------------
__global__ __launch_bounds__(64) void prep_kernel(const float* __restrict__ p,
                                                  const float* __restrict__ w,  // (23,6,3)
                                                  float* __restrict__ G)        // 23*128*128
{
    __shared__ float2 st[QDIM][QDIM];   // st[col][amp] (32 KB)
    const int col = threadIdx.x;

    for (int l = 0; l < NLAY; ++l) {
        for (int i = 0; i < QDIM; ++i) st[col][i] = make_float2(0.f, 0.f);
        st[col][col] = make_float2(1.f, 0.f);

        // ---- RZ(p[q]) on every qubit (layers 0..21 only) ----
        if (l < NLAY - 1) {
            for (int q = 0; q < NQ; ++q) {
                float s, c;
                __sincosf(0.5f * p[q], &s, &c);
                const int bpos = 5 - q;                  // qubit q = bit (5-q)
                for (int i = 0; i < QDIM; ++i) {
                    float sg = ((i >> bpos) & 1) ? s : -s;   // bit0: c-is, bit1: c+is
                    float2 v = st[col][i];
                    st[col][i] = make_float2(c * v.x - sg * v.y, c * v.y + sg * v.x);
                }
            }
        }
        // ---- VQC: per qubit RX(w0) then RZ(w1) ----
        for (int q = 0; q < NQ; ++q) {
            const float t0 = w[(l * NQ + q) * 3 + 0];
            const float t1 = w[(l * NQ + q) * 3 + 1];
            const int bpos = 5 - q;
            float s, c;
            __sincosf(0.5f * t0, &s, &c);
            for (int pp = 0; pp < QDIM / 2; ++pp) {
                int i0 = ((pp >> bpos) << (bpos + 1)) | (pp & ((1 << bpos) - 1));
                int i1 = i0 | (1 << bpos);
                float2 a = st[col][i0], b = st[col][i1];
                st[col][i0] = make_float2(c * a.x + s * b.y, c * a.y - s * b.x);
                st[col][i1] = make_float2(s * a.y + c * b.x, -s * a.x + c * b.y);
            }
            float sz, cz;
            __sincosf(0.5f * t1, &sz, &cz);
            for (int i = 0; i < QDIM; ++i) {
                float sg = ((i >> bpos) & 1) ? sz : -sz;
                float2 v = st[col][i];
                st[col][i] = make_float2(cz * v.x - sg * v.y, cz * v.y + sg * v.x);
            }
        }
        // ---- CRX chain: q=0..5, control=5-q, target=(6-q)%6 ----
        for (int q = 0; q < NQ; ++q) {
            const int cq = NQ - q - 1;
            const int tq = (NQ - q) % NQ;
            const float th = w[(l * NQ + q) * 3 + 2];
            float s, c;
            __sincosf(0.5f * th, &s, &c);
            const int bc = 1 << (5 - cq), bt = 1 << (5 - tq);
            for (int i = 0; i < QDIM; ++i) {
                if ((i & bc) && !(i & bt)) {
                    int i1 = i | bt;
                    float2 a = st[col][i], b = st[col][i1];
                    st[col][i]  = make_float2(c * a.x + s * b.y, c * a.y - s * b.x);
                    st[col][i1] = make_float2(s * a.y + c * b.x, -s * a.x + c * b.y);
                }
            }
        }
        // ---- emit swizzled real operator rows k=2col, 2col+1 ----
        float* Gl = G + l * 128 * 128;
        const int k0 = 2 * col, k1 = 2 * col + 1;
        for (int b = 0; b < QDIM; ++b) {
            float2 f = st[col][b];                 // F[b][col]
            Gl[gbidx(k0, 2 * b + 0)] =  f.x;       // Re
            Gl[gbidx(k0, 2 * b + 1)] =  f.y;       // Im
            Gl[gbidx(k1, 2 * b + 0)] = -f.y;
            Gl[gbidx(k1, 2 * b + 1)] =  f.x;
        }
        __syncthreads();
    }
}

// ---------------------------------------------------------------------------
// Main kernel: 256 threads = 8 waves; wave evolves 16 elements.
// Dynamic LDS = 64KB (swizzled G) + 64KB (8 wave states) = 128 KB.
// ---------------------------------------------------------------------------
__global__ __launch_bounds__(256) void vqc_main(const float* __restrict__ G,
                                                const float* __restrict__ x,
                                                float* __restrict__ out,
                                                int nelem)
{
    extern __shared__ float smem[];
    float* Gs = smem;                 // 128*128 swizzled
    float* Ss = smem + 128 * 128;     // 8 waves * 16*128

    const int tid  = threadIdx.x;
    const int w    = tid >> 5;
    const int lane = tid & 31;
    const int m    = lane & 15;       // element row / N within tile
    const int hk   = lane >> 4;       // lane half -> K-pair select

    float* Sw   = Ss + w * (16 * 128);
    float* Srow = Sw + m * 128;

    const int elem = blockIdx.x * 128 + w * 16 + m;
    const int ec   = elem < nelem ? elem : nelem - 1;

    // Data-gate matrix A = RY(x1)*RX(x0)
    float x0 = x[2 * ec], x1 = x[2 * ec + 1];
    float cs, ss, c2, s2;
    __sincosf(0.5f * x0, &ss, &cs);
    __sincosf(0.5f * x1, &s2, &c2);
    const float a00r =  c2 * cs, a00i =  s2 * ss;
    const float a01r = -s2 * cs, a01i = -c2 * ss;
    const float a10r =  s2 * cs, a10i = -c2 * ss;
    const float a11r =  c2 * cs, a11i = -s2 * ss;

    // |0...0>
    for (int j = 0; j < 64; ++j) Srow[hk * 64 + j] = 0.f;
    if (hk == 0) Srow[0] = 1.0f;

    // Register staging for the G pipeline: 16 float4 = this thread's 64 floats.
    float4 gr[16];
    {
        const float4* Gl0 = (const float4*)G;      // layer 0, swizzled
        #pragma unroll
        for (int i = 0; i < 16; ++i) gr[i] = Gl0[i * 256 + tid];
    }

    for (int l = 0; l < NLAY; ++l) {
        __syncthreads();                            // all waves done reading Gs
        // dump staged G_l registers into LDS
        float4* Gs4 = (float4*)Gs;
        #pragma unroll
        for (int i = 0; i < 16; ++i) Gs4[i * 256 + tid] = gr[i];
        // issue global loads for G_{l+1}; they land during the GEMM below
        if (l + 1 < NLAY) {
            const float4* Gln = (const float4*)(G + (l + 1) * 16384);
            #pragma unroll
            for (int i = 0; i < 16; ++i) gr[i] = Gln[i * 256 + tid];
        }
        __syncthreads();                            // Gs fill visible block-wide

        // ---- GEMM: new_state(16x128) = state(16x128) x G_l(128x128) ----
        v8f acc[8];
        #pragma unroll
        for (int nt = 0; nt < 8; ++nt)
            acc[nt] = (v8f){0.f, 0.f, 0.f, 0.f, 0.f, 0.f, 0.f, 0.f};

        const v2f*    Av  = (const v2f*)Srow;       // A: ds_load_b64
        const float4* GBv = (const float4*)Gs;      // B: ds_load_b128 (2 N-tiles)
        for (int kt = 0; kt < 32; ++kt) {
            v2f a = Av[kt * 2 + hk];                // (M=m, K=4kt+2hk+{0,1})
            const int kb = ((kt * 2 + hk) * 4) * 16 + m;
            #pragma unroll
            for (int nt2 = 0; nt2 < 4; ++nt2) {
                float4 bv = GBv[kb + nt2 * 16];
                v2f be = {bv.x, bv.y};              // N-tile 2*nt2
                v2f bo = {bv.z, bv.w};              // N-tile 2*nt2+1
                acc[2 * nt2 + 0] = __builtin_amdgcn_wmma_f32_16x16x4_f32(
                    false, a, false, be, (short)0, acc[2 * nt2 + 0], false, false);
                acc[2 * nt2 + 1] = __builtin_amdgcn_wmma_f32_16x16x4_f32(
                    false, a, false, bo, (short)0, acc[2 * nt2 + 1], false, false);
            }
        }

        // ---- write back D (VGPR r -> row r+8*hk, col nt*16+m) ----
        #pragma unroll
        for (int nt = 0; nt < 8; ++nt) {
            #pragma unroll
            for (int r = 0; r < 8; ++r)
                Sw[(8 * hk + r) * 128 + nt * 16 + m] = acc[nt][r];
        }
        asm volatile("s_wait_dscnt 0x0" ::: "memory");   // cross-lane visibility

        // ---- per-element data gates (layers 0..21) ----
        if (l < NLAY - 1) {
            #pragma unroll
            for (int q = 0; q < NQ; ++q) {
                const int bpos = 5 - q;
                for (int t = 0; t < 16; ++t) {       // lane-half splits 32 pairs
                    int pp = (hk << 4) | t;
                    int i0 = ((pp >> bpos) << (bpos + 1)) | (pp & ((1 << bpos) - 1));
                    int i1 = i0 | (1 << bpos);
                    float p0r = Srow[2 * i0], p0i = Srow[2 * i0 + 1];
                    float p1r = Srow[2 * i1], p1i = Srow[2 * i1 + 1];
                    float n0r = a00r * p0r - a00i * p0i + a01r * p1r - a01i * p1i;
                    float n0i = a00r * p0i + a00i * p0r + a01r * p1i + a01i * p1r;
                    float n1r = a10r * p0r - a10i * p0i + a11r * p1r - a11i * p1i;
                    float n1i = a10r * p0i + a10i * p0r + a11r * p1i + a11i * p1r;
                    Srow[2 * i0] = n0r; Srow[2 * i0 + 1] = n0i;
                    Srow[2 * i1] = n1r; Srow[2 * i1 + 1] = n1i;
                }
                asm volatile("s_wait_dscnt 0x0" ::: "memory");
            }
        }
    }

    // ---- readout: z = sum_{a<32}|psi_a|^2 - sum_{a>=32}|psi_a|^2 ----
    float part = 0.f;
    for (int j = 0; j < 16; ++j) {
        float4 v = ((float4*)Srow)[hk * 16 + j];
        part += v.x * v.x + v.y * v.y + v.z * v.z + v.w * v.w;
    }
    float other = __shfl_xor(part, 16, 32);
    if (hk == 0 && elem < nelem) {
        float z = part - other;
        out[elem] = 0.5f * (z + 1.0f);
    }
}

// ---------------------------------------------------------------------------
extern "C" void kernel_launch(void* const* d_in, const int* in_sizes, int n_in,
                              void* d_out, int out_size, void* d_ws, size_t ws_size,
                              hipStream_t stream) {
    const float* p   = (const float*)d_in[0];   // (6,)
    const float* x   = (const float*)d_in[1];   // (20,64,64,2) -> flat pairs
    const float* wts = (const float*)d_in[2];   // (23,6,3)
    float* out = (float*)d_out;
    float* G   = (float*)d_ws;                  // 23 * 128*128 f32 = 1.47 MB

    prep_kernel<<<1, 64, 0, stream>>>(p, wts, G);

    const int nelem   = in_sizes[1] / 2;        // 81920
    const int nblocks = (nelem + 127) / 128;
    const size_t lds  = (size_t)(128 * 128 + 8 * 16 * 128) * sizeof(float); // 128 KB
    vqc_main<<<nblocks, 256, lds, stream>>>(G, x, out, nelem);
}